// GIN_78091095376035
// MI455X (gfx1250) — compile-verified
//
#include <hip/hip_runtime.h>

// GIN forward for MI455X (gfx1250, wave32).
// Dominant cost: edge scatter-gather (~1.2 GB total @ 23.3 TB/s). GEMMs (3.3 GF)
// run on the exact-precision f32 WMMA path (v_wmma_f32_16x16x4_f32).

#define HD      64      // feature / hidden dim
#define NGRAPH  64
#define OUTD    10
#define NLAYERS 4       // message-passing layers (NUM_LAYERS-1)

typedef __attribute__((ext_vector_type(2))) float v2f;
typedef __attribute__((ext_vector_type(8))) float v8f;

// ---------------------------------------------------------------- utilities
__global__ void zero_kernel(float* __restrict__ p, int n) {
  for (int i = blockIdx.x * blockDim.x + threadIdx.x; i < n;
       i += gridDim.x * blockDim.x)
    p[i] = 0.0f;
}

__global__ void copy4_kernel(float4* __restrict__ dst, const float4* __restrict__ src, int n4) {
  for (int i = blockIdx.x * blockDim.x + threadIdx.x; i < n4;
       i += gridDim.x * blockDim.x)
    dst[i] = src[i];
}

// graph sum-pool of a dense feature matrix (used for the raw input x)
__global__ void pool_graph_kernel(const float* __restrict__ h, const int* __restrict__ gids,
                                  float* __restrict__ gpool, int n) {
  const int total = n * HD;
  for (int i = blockIdx.x * blockDim.x + threadIdx.x; i < total;
       i += gridDim.x * blockDim.x) {
    const int node = i >> 6, d = i & 63;
    atomicAdd(&gpool[gids[node] * HD + d], h[i]);
  }
}

// ------------------------------------------------- COO scatter-sum (SpMM)
// 16 consecutive lanes handle one edge: coalesced 256B row gather, float4 chunks.
__global__ void agg_edge_kernel(const float* __restrict__ h, const int* __restrict__ dst,
                                const int* __restrict__ src, float* __restrict__ P, int E) {
  const int total = E * 16;
  for (int t = blockIdx.x * blockDim.x + threadIdx.x; t < total;
       t += gridDim.x * blockDim.x) {
    const int e = t >> 4, c = t & 15;
    const int s = src[e], d = dst[e];
    const float4 v = *(const float4*)(h + (size_t)s * HD + c * 4);
    float* q = P + (size_t)d * HD + c * 4;
    atomicAdd(q + 0, v.x);
    atomicAdd(q + 1, v.y);
    atomicAdd(q + 2, v.z);
    atomicAdd(q + 3, v.w);
  }
}

// --------------------------------------------- WMMA GEMM: Z = A @ W + bias
// A: N x 64 (row-major), W: 64 x 64 (row-major), Z: N x 64.
// One wave computes a 16-row x 64-col slab: 4 N-tiles x 16 K-steps of
// v_wmma_f32_16x16x4_f32 (exact f32). Epilogue accumulates per-column
// sum / sum-of-squares for training-mode BatchNorm.
__global__ void gemm64_wmma_kernel(const float* __restrict__ A, const float* __restrict__ W,
                                   const float* __restrict__ bias, float* __restrict__ Z,
                                   float* __restrict__ colsum, float* __restrict__ colsumsq,
                                   int N) {
  const int wid  = threadIdx.x >> 5;
  const int lane = threadIdx.x & 31;
  const int r0   = (blockIdx.x * 8 + wid) * 16;   // 8 waves per 256-thread block
  if (r0 >= N) return;                            // wave-uniform guard (EXEC stays all-1)

  const int m   = lane & 15;                      // M index within tile
  const int hi  = lane >> 4;                      // lane-half: selects K pair {0,1}/{2,3}
  const int row = r0 + m;

  // A fragments for all 16 K-chunks (ISA 32-bit A 16x4 layout):
  // lanes 0-15 hold (K=4kc, 4kc+1); lanes 16-31 hold (K=4kc+2, 4kc+3).
  v2f a[16];
  const float* arow = A + (size_t)row * HD + hi * 2;
#pragma unroll
  for (int kc = 0; kc < 16; ++kc) {
    const float2 t = *(const float2*)(arow + 4 * kc);
    a[kc].x = t.x;
    a[kc].y = t.y;
  }

#pragma unroll
  for (int j = 0; j < 4; ++j) {
    const int col = j * 16 + m;
    const float bv = bias[col];
    v8f acc = {bv, bv, bv, bv, bv, bv, bv, bv};   // bias folded into C
#pragma unroll
    for (int kc = 0; kc < 16; ++kc) {
      const int k = 4 * kc + 2 * hi;
      v2f b;                                      // B 4x16: lane = N, half/comp = K
      b.x = W[(k + 0) * HD + col];
      b.y = W[(k + 1) * HD + col];
      acc = __builtin_amdgcn_wmma_f32_16x16x4_f32(
          /*neg_a=*/false, a[kc], /*neg_b=*/false, b,
          /*c_mod=*/(short)0, acc, /*reuse_a=*/false, /*reuse_b=*/false);
    }
    // C/D layout: VGPR i -> row r0 + 8*hi + i, col = j*16 + m
    float s = 0.0f, s2 = 0.0f;
#pragma unroll
    for (int i = 0; i < 8; ++i) {
      const float v = acc[i];
      Z[(size_t)(r0 + 8 * hi + i) * HD + col] = v;
      s += v;
      s2 += v * v;
    }
    atomicAdd(&colsum[col], s);
    atomicAdd(&colsumsq[col], s2);
  }
}

// ---------------------------------- BatchNorm(train, biased var) + ReLU
// Optionally fuses the graph sum-pool of the produced activations.
__global__ void bnrelu_kernel(const float* __restrict__ Z, const float* __restrict__ cs,
                              const float* __restrict__ cs2, const float* __restrict__ g,
                              const float* __restrict__ b, float* __restrict__ H, int N,
                              const int* __restrict__ gids, float* __restrict__ gpool) {
  const int total = N * HD;
  const float invN = 1.0f / (float)N;
  for (int i = blockIdx.x * blockDim.x + threadIdx.x; i < total;
       i += gridDim.x * blockDim.x) {
    const int d = i & 63, node = i >> 6;
    const float mu   = cs[d] * invN;
    const float var  = cs2[d] * invN - mu * mu;
    const float istd = rsqrtf(var + 1e-5f);
    float h = g[d] * (Z[i] - mu) * istd + b[d];
    h = h > 0.0f ? h : 0.0f;
    H[i] = h;
    if (gpool) atomicAdd(&gpool[gids[node] * HD + d], h);
  }
}

// ------------------------------------------------------- per-graph readout
struct PredPtrs {
  const float* W[NLAYERS + 1];
  const float* b[NLAYERS + 1];
};

__global__ void readout_kernel(const float* __restrict__ gpool, PredPtrs p,
                               float* __restrict__ out) {
  const int t = blockIdx.x * blockDim.x + threadIdx.x;
  if (t >= NGRAPH * OUTD) return;
  const int gr = t / OUTD, o = t % OUTD;
  float acc = 0.0f;
  for (int l = 0; l <= NLAYERS; ++l) {
    acc += p.b[l][o];
    const float* gp = gpool + (size_t)l * NGRAPH * HD + (size_t)gr * HD;
    const float* w = p.W[l];
#pragma unroll 8
    for (int d = 0; d < HD; ++d) acc += gp[d] * w[d * OUTD + o];
  }
  out[t] = acc;
}

// ------------------------------------------------------------------- host
extern "C" void kernel_launch(void* const* d_in, const int* in_sizes, int n_in,
                              void* d_out, int out_size, void* d_ws, size_t ws_size,
                              hipStream_t stream) {
  (void)n_in; (void)out_size; (void)ws_size;

  const float* x    = (const float*)d_in[0];
  const int*   ei   = (const int*)d_in[1];    // [2,E] flat: dst rows then src rows
  const int*   gids = (const int*)d_in[2];

  const int N = in_sizes[0] / HD;
  const int E = in_sizes[1] / 2;

  const float *lin1W[NLAYERS], *lin1b[NLAYERS], *bn1g[NLAYERS], *bn1b[NLAYERS];
  const float *lin2W[NLAYERS], *lin2b[NLAYERS], *bng[NLAYERS], *bnb[NLAYERS];
  const float *predW[NLAYERS + 1], *predb[NLAYERS + 1];

  int idx = 3;
  if (in_sizes[3] == HD * HD) {
    // dict-insertion flatten: mlps{lin1.W,lin1.b,bn1_g,bn1_b,lin2.W,lin2.b} x4,
    // bns{g,b} x4, preds{W,b} x5
    for (int l = 0; l < NLAYERS; ++l) {
      lin1W[l] = (const float*)d_in[idx++];
      lin1b[l] = (const float*)d_in[idx++];
      bn1g[l]  = (const float*)d_in[idx++];
      bn1b[l]  = (const float*)d_in[idx++];
      lin2W[l] = (const float*)d_in[idx++];
      lin2b[l] = (const float*)d_in[idx++];
    }
    for (int l = 0; l < NLAYERS; ++l) {
      bng[l] = (const float*)d_in[idx++];
      bnb[l] = (const float*)d_in[idx++];
    }
    for (int l = 0; l <= NLAYERS; ++l) {
      predW[l] = (const float*)d_in[idx++];
      predb[l] = (const float*)d_in[idx++];
    }
  } else {
    // jax pytree (sorted-key) flatten: bns{b,g} x4,
    // mlps{bn1_b,bn1_g,lin1.W,lin1.b,lin2.W,lin2.b} x4, preds{W,b} x5
    for (int l = 0; l < NLAYERS; ++l) {
      bnb[l] = (const float*)d_in[idx++];
      bng[l] = (const float*)d_in[idx++];
    }
    for (int l = 0; l < NLAYERS; ++l) {
      bn1b[l]  = (const float*)d_in[idx++];
      bn1g[l]  = (const float*)d_in[idx++];
      lin1W[l] = (const float*)d_in[idx++];
      lin1b[l] = (const float*)d_in[idx++];
      lin2W[l] = (const float*)d_in[idx++];
      lin2b[l] = (const float*)d_in[idx++];
    }
    for (int l = 0; l <= NLAYERS; ++l) {
      predW[l] = (const float*)d_in[idx++];
      predb[l] = (const float*)d_in[idx++];
    }
  }

  // workspace layout (floats): H | P | Z | T | stats[4][4][64] | gpool[5][64*64]
  const size_t N64 = (size_t)N * HD;
  float* ws    = (float*)d_ws;
  float* H     = ws;
  float* P     = ws + N64;
  float* Z     = ws + 2 * N64;
  float* T     = ws + 3 * N64;
  float* stats = ws + 4 * N64;                 // per layer: sum1|sq1|sum2|sq2
  float* gpool = stats + NLAYERS * 4 * HD;
  const int statsN = NLAYERS * 4 * HD + (NLAYERS + 1) * NGRAPH * HD;

  const int elemGrid = (N * HD + 255) / 256;
  const int gemmGrid = (N + 127) / 128;        // 8 waves/block * 16 rows/wave

  zero_kernel<<<(statsN + 255) / 256, 256, 0, stream>>>(stats, statsN);
  pool_graph_kernel<<<elemGrid, 256, 0, stream>>>(x, gids, gpool, N);

  const float* hcur = x;
  for (int l = 0; l < NLAYERS; ++l) {
    // pooled = h (self-loop) + scatter-sum over edges
    copy4_kernel<<<(N * HD / 4 + 255) / 256, 256, 0, stream>>>(
        (float4*)P, (const float4*)hcur, N * HD / 4);
    agg_edge_kernel<<<(E * 16 + 255) / 256, 256, 0, stream>>>(hcur, ei, ei + E, P, E);

    float* st = stats + l * 4 * HD;
    gemm64_wmma_kernel<<<gemmGrid, 256, 0, stream>>>(P, lin1W[l], lin1b[l], Z,
                                                     st, st + HD, N);
    bnrelu_kernel<<<elemGrid, 256, 0, stream>>>(Z, st, st + HD, bn1g[l], bn1b[l],
                                                T, N, nullptr, nullptr);
    gemm64_wmma_kernel<<<gemmGrid, 256, 0, stream>>>(T, lin2W[l], lin2b[l], Z,
                                                     st + 2 * HD, st + 3 * HD, N);
    bnrelu_kernel<<<elemGrid, 256, 0, stream>>>(Z, st + 2 * HD, st + 3 * HD, bng[l],
                                                bnb[l], H, N, gids,
                                                gpool + (size_t)(l + 1) * NGRAPH * HD);
    hcur = H;
  }

  PredPtrs pp;
  for (int l = 0; l <= NLAYERS; ++l) { pp.W[l] = predW[l]; pp.b[l] = predb[l]; }
  readout_kernel<<<(NGRAPH * OUTD + 255) / 256, 256, 0, stream>>>(gpool, pp,
                                                                  (float*)d_out);
}